// HungarianMatcher_75771813036148
// MI455X (gfx1250) — compile-verified
//
#include <hip/hip_runtime.h>

typedef float v2f __attribute__((ext_vector_type(2)));
typedef float v8f __attribute__((ext_vector_type(8)));

#define W_BBOX 5.0f
#define W_GIOU 2.0f
#define EPS_F 1e-6f

constexpr int NQ = 900;
constexpr int NC = 256;
constexpr int NT = 128;
constexpr int QT = 16;                               // queries per wave tile
constexpr int WPB = 2;                               // waves per block
constexpr int NTILES = (NQ + QT - 1) / QT;           // 57
constexpr int BLKS_PER_B = (NTILES + WPB - 1) / WPB; // 29
constexpr int PSTRIDE = 260;                         // prob row stride (bank-conflict padding)

__global__ __launch_bounds__(WPB * 32)
void matcher_cost_kernel(const float* __restrict__ logits,
                         const float* __restrict__ pboxes,
                         const int*   __restrict__ tlabels,
                         const float* __restrict__ tboxes,
                         float* __restrict__ out)
{
    __shared__ float s_prob[WPB][QT][PSTRIDE];   // normalized softmax probs
    __shared__ float s_pbox[WPB][QT][4];         // pred cxcywh
    __shared__ float s_pxyxy[WPB][QT][4];        // pred xyxy
    __shared__ float s_parea[WPB][QT];           // pred area
    __shared__ int   s_tlbl[NT];
    __shared__ float s_tbox[NT][4];              // tgt cxcywh
    __shared__ float s_txyxy[NT][4];             // tgt xyxy
    __shared__ float s_tarea[NT];

    const int b    = blockIdx.x / BLKS_PER_B;
    const int lb   = blockIdx.x % BLKS_PER_B;
    const int tid  = threadIdx.x;
    const int w    = tid >> 5;
    const int lane = tid & 31;
    const int half = lane >> 4;
    const int ln16 = lane & 15;
    const int tile = lb * WPB + w;
    const int q0   = tile * QT;

    // ---- stage targets (block-cooperative) ----
    for (int t = tid; t < NT; t += WPB * 32) {
        s_tlbl[t] = tlabels[b * NT + t];
        const float* tb = tboxes + ((size_t)b * NT + t) * 4;
        float cx = tb[0], cy = tb[1], tw = tb[2], th = tb[3];
        s_tbox[t][0] = cx; s_tbox[t][1] = cy; s_tbox[t][2] = tw; s_tbox[t][3] = th;
        float x1 = cx - 0.5f * tw, y1 = cy - 0.5f * th;
        float x2 = cx + 0.5f * tw, y2 = cy + 0.5f * th;
        s_txyxy[t][0] = x1; s_txyxy[t][1] = y1; s_txyxy[t][2] = x2; s_txyxy[t][3] = y2;
        s_tarea[t] = fmaxf(x2 - x1, 0.0f) * fmaxf(y2 - y1, 0.0f);
    }

    // ---- stage pred boxes (per wave, lanes 0..15) ----
    if (lane < QT) {
        int q = q0 + lane; if (q > NQ - 1) q = NQ - 1;
        const float* pb = pboxes + ((size_t)b * NQ + q) * 4;
        float cx = pb[0], cy = pb[1], pw = pb[2], ph = pb[3];
        s_pbox[w][lane][0] = cx; s_pbox[w][lane][1] = cy;
        s_pbox[w][lane][2] = pw; s_pbox[w][lane][3] = ph;
        float x1 = cx - 0.5f * pw, y1 = cy - 0.5f * ph;
        float x2 = cx + 0.5f * pw, y2 = cy + 0.5f * ph;
        s_pxyxy[w][lane][0] = x1; s_pxyxy[w][lane][1] = y1;
        s_pxyxy[w][lane][2] = x2; s_pxyxy[w][lane][3] = y2;
        s_parea[w][lane] = fmaxf(x2 - x1, 0.0f) * fmaxf(y2 - y1, 0.0f);
    }

    // ---- softmax: per-row max, sum(exp), stage normalized probs ----
    for (int qi = 0; qi < QT; ++qi) {
        int q = q0 + qi; if (q > NQ - 1) q = NQ - 1;
        const float* row = logits + ((size_t)b * NQ + q) * NC;
        float v[8], e[8];
#pragma unroll
        for (int i = 0; i < 8; ++i) v[i] = row[lane + 32 * i];
        float m = v[0];
#pragma unroll
        for (int i = 1; i < 8; ++i) m = fmaxf(m, v[i]);
#pragma unroll
        for (int off = 16; off >= 1; off >>= 1) m = fmaxf(m, __shfl_xor(m, off, 32));
        float s = 0.0f;
#pragma unroll
        for (int i = 0; i < 8; ++i) { e[i] = __expf(v[i] - m); s += e[i]; }
#pragma unroll
        for (int off = 16; off >= 1; off >>= 1) s += __shfl_xor(s, off, 32);
        float inv = 1.0f / s;
#pragma unroll
        for (int i = 0; i < 8; ++i) s_prob[w][qi][lane + 32 * i] = e[i] * inv;
    }

    __syncthreads();

    // ---- one-hot gather via f32 WMMA: cost_cls tile = P(16xC) x onehot(Cx128) ----
    int lbl[8];
#pragma unroll
    for (int t = 0; t < 8; ++t) lbl[t] = s_tlbl[t * 16 + ln16];

    v8f acc[8];
#pragma unroll
    for (int t = 0; t < 8; ++t) acc[t] = (v8f){0.f,0.f,0.f,0.f,0.f,0.f,0.f,0.f};

    const int kh = 2 * half;
    for (int k0 = 0; k0 < NC; k0 += 4) {
        const int kk = k0 + kh;
        // A fragment: lane holds P[row=ln16][kk], P[row=ln16][kk+1]
        v2f a = *(const v2f*)&s_prob[w][ln16][kk];
#pragma unroll
        for (int t = 0; t < 8; ++t) {
            v2f bf;
            bf[0] = (lbl[t] == kk)     ? 1.0f : 0.0f;
            bf[1] = (lbl[t] == kk + 1) ? 1.0f : 0.0f;
            acc[t] = __builtin_amdgcn_wmma_f32_16x16x4_f32(
                false, a, false, bf, (short)0, acc[t], false, false);
        }
    }

    // ---- finalize: -prob + 5*L1 - 2*GIoU, store [b][q][t] ----
#pragma unroll
    for (int t = 0; t < 8; ++t) {
        const int tt = t * 16 + ln16;
        float tcx = s_tbox[tt][0], tcy = s_tbox[tt][1];
        float twd = s_tbox[tt][2], tht = s_tbox[tt][3];
        float tx1 = s_txyxy[tt][0], ty1 = s_txyxy[tt][1];
        float tx2 = s_txyxy[tt][2], ty2 = s_txyxy[tt][3];
        float ta  = s_tarea[tt];
#pragma unroll
        for (int r = 0; r < 8; ++r) {
            const int qi = r + 8 * half;
            const int q  = q0 + qi;
            float pcx = s_pbox[w][qi][0], pcy = s_pbox[w][qi][1];
            float pwd = s_pbox[w][qi][2], pht = s_pbox[w][qi][3];
            float px1 = s_pxyxy[w][qi][0], py1 = s_pxyxy[w][qi][1];
            float px2 = s_pxyxy[w][qi][2], py2 = s_pxyxy[w][qi][3];
            float pa  = s_parea[w][qi];

            float prob = acc[t][r];

            float l1 = fabsf(pcx - tcx) + fabsf(pcy - tcy) +
                       fabsf(pwd - twd) + fabsf(pht - tht);

            float ltx = fmaxf(px1, tx1), lty = fmaxf(py1, ty1);
            float rbx = fminf(px2, tx2), rby = fminf(py2, ty2);
            float iw = fmaxf(rbx - ltx, 0.0f), ih = fmaxf(rby - lty, 0.0f);
            float inter = iw * ih;
            float uni = pa + ta - inter + EPS_F;
            float iou = inter / uni;
            float ex1 = fminf(px1, tx1), ey1 = fminf(py1, ty1);
            float ex2 = fmaxf(px2, tx2), ey2 = fmaxf(py2, ty2);
            float ew = fmaxf(ex2 - ex1, 0.0f), eh = fmaxf(ey2 - ey1, 0.0f);
            float enc = ew * eh + EPS_F;
            float giou = iou - (enc - uni) / enc;

            float cost = -prob + W_BBOX * l1 - W_GIOU * giou;

            if (q < NQ) out[((size_t)b * NQ + q) * NT + tt] = cost;
        }
    }
}

extern "C" void kernel_launch(void* const* d_in, const int* in_sizes, int n_in,
                              void* d_out, int out_size, void* d_ws, size_t ws_size,
                              hipStream_t stream) {
    const float* pred_logits = (const float*)d_in[0];
    const float* pred_boxes  = (const float*)d_in[1];
    const int*   tgt_labels  = (const int*)d_in[2];
    const float* tgt_boxes   = (const float*)d_in[3];
    float* out = (float*)d_out;

    const int Bn = in_sizes[2] / NT;  // batch count (64)
    dim3 grid(Bn * BLKS_PER_B);
    dim3 block(WPB * 32);
    matcher_cost_kernel<<<grid, block, 0, stream>>>(pred_logits, pred_boxes,
                                                    tgt_labels, tgt_boxes, out);
}